// NT_Xent_83863531421802
// MI455X (gfx1250) — compile-verified
//
#include <hip/hip_runtime.h>
#include <math.h>

typedef float v2f __attribute__((ext_vector_type(2)));
typedef float v8f __attribute__((ext_vector_type(8)));

#define DIMS 128
#define INV_T 2.0f      /* 1 / TEMPERATURE, TEMPERATURE = 0.5 */
#define EPS_CS 1e-8f

// One wave (32 lanes) handles 16 (x_i, x_j) pairs.
// 256 threads/block = 8 waves = 128 pairs per block.
__global__ __launch_bounds__(256) void ntxent_diag_kernel(
    const float* __restrict__ xi, const float* __restrict__ xj,
    float* __restrict__ partial, int B) {
  __shared__ float red[256];

  const int tid      = threadIdx.x;
  const int lane     = tid & 31;
  const int gwave    = blockIdx.x * 8 + (tid >> 5);
  const int pairBase = gwave * 16;

  float contrib = 0.0f;

  if (pairBase + 16 <= B) {  // wave-uniform guard: EXEC all-1s for WMMA
    // f32 16x16x4 A-fragment: lane = row (mod 16), K = 2*(lane/16) + {0,1}.
    // Identical layout serves as the B-fragment (column-per-lane) for X*X^T.
    const int row  = pairBase + (lane & 15);
    const int koff = (lane >> 4) * 2;
    const float* pi = xi + (size_t)row * DIMS + koff;
    const float* pj = xj + (size_t)row * DIMS + koff;

    v8f cii = {};  // diag -> ||x_i||^2
    v8f cjj = {};  // diag -> ||x_j||^2
    v8f cij = {};  // diag -> x_i . x_j

#pragma unroll
    for (int k = 0; k < DIMS; k += 4) {
      v2f ai = *(const v2f*)(pi + k);
      v2f aj = *(const v2f*)(pj + k);
      cii = __builtin_amdgcn_wmma_f32_16x16x4_f32(false, ai, false, ai,
                                                  (short)0, cii, false, false);
      cjj = __builtin_amdgcn_wmma_f32_16x16x4_f32(false, aj, false, aj,
                                                  (short)0, cjj, false, false);
      cij = __builtin_amdgcn_wmma_f32_16x16x4_f32(false, ai, false, aj,
                                                  (short)0, cij, false, false);
    }

    // C/D layout: VGPR v, lanes 0-15 -> (M=v, N=lane); lanes 16-31 -> (M=v+8, N=lane-16).
    // Diagonal (m,m): m in [0,8) at vgpr m / lane m; m in [8,16) at vgpr m-8 / lane m+16.
    const bool ownLo = (lane < 8);
    const bool ownHi = (lane >= 24);
    if (ownLo || ownHi) {
      const int idx = ownLo ? lane : (lane - 24);
      const float dii = cii[idx];
      const float djj = cjj[idx];
      const float dij = cij[idx];

      const float ni = fmaxf(sqrtf(dii), EPS_CS);
      const float nj = fmaxf(sqrtf(djj), EPS_CS);

      const float p    = (dij / (ni * nj)) * INV_T;  // sim[r, r+B] == sim[r+B, r]
      const float negI = (dii / (ni * ni)) * INV_T;  // sim[r, r]       (faithful "negative")
      const float negJ = (djj / (nj * nj)) * INV_T;  // sim[r+B, r+B]

      // -log_softmax([p, n])[0] = logsumexp(p, n) - p  (max-stabilized)
      const float m1 = fmaxf(p, negI);
      const float l1 = m1 + logf(expf(p - m1) + expf(negI - m1));
      const float m2 = fmaxf(p, negJ);
      const float l2 = m2 + logf(expf(p - m2) + expf(negJ - m2));
      contrib = (l1 - p) + (l2 - p);  // rows r and r+B
    }
  }

  // Deterministic fixed-shape block tree reduction.
  red[tid] = contrib;
  __syncthreads();
#pragma unroll
  for (int s = 128; s > 0; s >>= 1) {
    if (tid < s) red[tid] += red[tid + s];
    __syncthreads();
  }
  if (tid == 0) partial[blockIdx.x] = red[0];
}

__global__ void ntxent_finalize(const float* __restrict__ partial,
                                float* __restrict__ out, int nparts, int B) {
  if (threadIdx.x == 0 && blockIdx.x == 0) {
    float s = 0.0f;
    for (int i = 0; i < nparts; ++i) s += partial[i];  // fixed order: deterministic
    out[0] = s / (float)(2 * B);                       // N = 2B
  }
}

extern "C" void kernel_launch(void* const* d_in, const int* in_sizes, int n_in,
                              void* d_out, int out_size, void* d_ws, size_t ws_size,
                              hipStream_t stream) {
  const float* xi = (const float*)d_in[0];
  const float* xj = (const float*)d_in[1];
  float* out = (float*)d_out;
  float* partial = (float*)d_ws;

  const int B = in_sizes[0] / DIMS;          // 4096
  const int blocks = (B + 127) / 128;        // 128 pairs per block

  ntxent_diag_kernel<<<blocks, 256, 0, stream>>>(xi, xj, partial, B);
  ntxent_finalize<<<1, 32, 0, stream>>>(partial, out, blocks, B);
}